// CapsuleLayer_84018150245170
// MI455X (gfx1250) — compile-verified
//
#include <hip/hip_runtime.h>
#include <hip/hip_bf16.h>
#include <math.h>

typedef __attribute__((ext_vector_type(2))) float v2f;
typedef __attribute__((ext_vector_type(8))) float v8f;

#define B_SZ   256
#define IC     1152
#define ID     8
#define NC     10
#define DC     16
#define NDCOL  (NC*DC)          // 160
#define ICHUNK 16
#define NCHUNKS (IC/ICHUNK)     // 72
#define EPSQ   1e-7f

// ---------------------------------------------------------------- squash ----
__device__ __forceinline__ float squash1(float s) {
    float sq = s * s;
    // sq/(1+sq)/sqrt(sq+eps) * s   (elementwise, matching the reference quirk)
    return (sq / ((1.0f + sq) * sqrtf(sq + EPSQ))) * s;
}

// ------------------------------------------------------------ zero s1 -------
__global__ void caps_zero_s1(float* __restrict__ s1) {
    int t = blockIdx.x * blockDim.x + threadIdx.x;
    if (t < B_SZ * NDCOL) s1[t] = 0.0f;
}

// ------------------------------------------------- u_hat GEMM via WMMA ------
// wave <-> (b-tile 0..15, n 0..9, i-chunk 0..71); per i: two v_wmma_f32_16x16x4_f32
// D tile = 16(b) x 16(d). Also accumulates sum_i of the tile and atomically adds
// it into s1 (routing iteration 0: s1 = sum_i u_hat, scaled by 0.1 later).
__global__ __launch_bounds__(256) void caps_uhat_wmma(
    const float* __restrict__ inp,   // [256,1152,8]
    const float* __restrict__ W,     // [10,1152,16,8]
    float* __restrict__ u_ws,        // [256,1152,10,16]
    float* __restrict__ s1)          // [256,160]
{
    const int wave = blockIdx.x * (blockDim.x >> 5) + (threadIdx.x >> 5);
    const int lane = threadIdx.x & 31;
    const int ic   = wave % NCHUNKS;
    const int n    = (wave / NCHUNKS) % NC;
    const int bt   = wave / (NCHUNKS * NC);   // 0..15
    const int m    = lane & 15;               // A: row M ; B: col d
    const int hi   = lane >> 4;               // selects K pair {0,1} vs {2,3}

    const float* aRow = inp + (size_t)(bt * 16 + m) * (IC * ID);
    const int i0 = ic * ICHUNK;

    v8f ssum = {0.f,0.f,0.f,0.f,0.f,0.f,0.f,0.f};

    for (int i = i0; i < i0 + ICHUNK; ++i) {
        const float* ap = aRow + i * ID + hi * 2;
        v2f a0 = *(const v2f*)(ap);          // K = hi*2, hi*2+1
        v2f a1 = *(const v2f*)(ap + 4);      // K = 4+hi*2, 5+hi*2
        const float* wp = W + (((size_t)(n * IC + i) * DC + m) * ID) + hi * 2;
        v2f b0 = *(const v2f*)(wp);
        v2f b1 = *(const v2f*)(wp + 4);

        v8f c = {0.f,0.f,0.f,0.f,0.f,0.f,0.f,0.f};
        c = __builtin_amdgcn_wmma_f32_16x16x4_f32(false, a0, false, b0,
                                                  (short)0, c, false, false);
        c = __builtin_amdgcn_wmma_f32_16x16x4_f32(false, a1, false, b1,
                                                  (short)0, c, false, false);
        ssum += c;

        // D layout: VGPR r holds row M = r + 8*hi, col N = lane%16
        #pragma unroll
        for (int r = 0; r < 8; ++r) {
            int brow = bt * 16 + r + 8 * hi;
            u_ws[(((size_t)brow * IC + i) * NC + n) * DC + m] = c[r];
        }
    }

    #pragma unroll
    for (int r = 0; r < 8; ++r) {
        int brow = bt * 16 + r + 8 * hi;
        atomicAdd(&s1[brow * NDCOL + n * DC + m], ssum[r]);
    }
}

// ---------------------------------------------------- routing iterations ----
// One 256-thread workgroup per batch element. Thread = (replica r=tid/16, d=tid%16);
// replica r owns i = r, r+16, ... (72 rows). Logits are always u_hat * vsum.
__global__ __launch_bounds__(256) void caps_routing(
    const float* __restrict__ u_ws,  // [256,1152,10,16]
    const float* __restrict__ s1,    // [256,160]
    float* __restrict__ out)         // [256,10,16,1]
{
    __shared__ float spart[16 * NDCOL];
    __shared__ float vsum[NDCOL];

    const int b   = blockIdx.x;
    const int tid = threadIdx.x;
    const int d   = tid & 15;
    const int rr  = tid >> 4;       // 0..15

    // --- routing iteration 0: b=0 -> c = 1/10 uniform -> s = 0.1 * sum_i u
    if (tid < NDCOL) {
        float s = 0.1f * s1[b * NDCOL + tid];
        vsum[tid] = squash1(s);      // vsum = v1
    }
    __syncthreads();

    const float* ub = u_ws + (size_t)b * IC * NDCOL;
    float vout = 0.0f;

    for (int it = 1; it < 3; ++it) {
        float vs[NC], sp[NC];
        #pragma unroll
        for (int n = 0; n < NC; ++n) { vs[n] = vsum[n * DC + d]; sp[n] = 0.f; }

        for (int i = rr; i < IC; i += 16) {
            const float* up = ub + (size_t)i * NDCOL + d;
            float u[NC], t[NC];
            float mx = -3.4e38f;
            #pragma unroll
            for (int n = 0; n < NC; ++n) {
                u[n] = up[n * DC];
                t[n] = u[n] * vs[n];
                mx = fmaxf(mx, t[n]);
            }
            float e[NC], Z = 0.f;
            #pragma unroll
            for (int n = 0; n < NC; ++n) { e[n] = __expf(t[n] - mx); Z += e[n]; }
            float invZ = 1.0f / Z;
            #pragma unroll
            for (int n = 0; n < NC; ++n) sp[n] += e[n] * invZ * u[n];
        }

        #pragma unroll
        for (int n = 0; n < NC; ++n) spart[rr * NDCOL + n * DC + d] = sp[n];
        __syncthreads();

        if (tid < NDCOL) {
            float s = 0.f;
            #pragma unroll
            for (int r2 = 0; r2 < 16; ++r2) s += spart[r2 * NDCOL + tid];
            float v = squash1(s);
            if (it == 1) vsum[tid] += v;   // vsum = v1 + v2 (logit accumulator)
            else         vout = v;         // v3 = final output
        }
        __syncthreads();
    }

    if (tid < NDCOL) out[b * NDCOL + tid] = vout;
}

// ---------------------------------------------------------------- launch ----
extern "C" void kernel_launch(void* const* d_in, const int* in_sizes, int n_in,
                              void* d_out, int out_size, void* d_ws, size_t ws_size,
                              hipStream_t stream) {
    const float* inp = (const float*)d_in[0];   // [256,1152,8]
    const float* W   = (const float*)d_in[1];   // [10,1152,16,8]
    float* out  = (float*)d_out;                // [256,10,16,1] flat
    float* u_ws = (float*)d_ws;                               // 47,185,920 f32
    float* s1   = u_ws + (size_t)B_SZ * IC * NC * DC;         // + 40,960 f32

    caps_zero_s1<<<(B_SZ * NDCOL + 255) / 256, 256, 0, stream>>>(s1);

    const int waves = 16 * NC * NCHUNKS;        // 11520 waves, 8 per block
    caps_uhat_wmma<<<waves / 8, 256, 0, stream>>>(inp, W, u_ws, s1);

    caps_routing<<<B_SZ, 256, 0, stream>>>(u_ws, s1, out);
}